// ProbabilisticChamferDistanceLoss_55198919688758
// MI455X (gfx1250) — compile-verified
//
#include <hip/hip_runtime.h>
#include <math.h>

typedef __attribute__((ext_vector_type(2))) float v2f;
typedef __attribute__((ext_vector_type(8))) float v8f;

// One wave owns a 16-row tile of query points and scans all target points in
// 16-column tiles. Squared distance comes straight out of the WMMA:
//   D = (-2*A) x B + C,  with C[v] = |q_row|^2 + |t_col|^2
// so d^2 = |q|^2 + |t|^2 - 2 q.t with zero epilogue FMAs.
//
// A (16x4 f32) layout: lanes 0-15 hold K=0,1 (x,y) of row=lane; lanes 16-31
// hold K=2,3 (z,0) of row=lane-16.  B (4x16) mirrors it with N=lane&15.
// D (16x16 f32) layout: VGPR v, lanes 0-15 -> row v, col lane; lanes 16-31 ->
// row v+8, col lane-16.
template <bool WEIGHT_BY_TARGET>
__global__ void __launch_bounds__(256)
chamfer_dir_kernel(const float* __restrict__ Q, int nq,
                   const float* __restrict__ T, int nt,
                   const float* __restrict__ prob,
                   float* __restrict__ partials) {
  const int lane = threadIdx.x & 31;
  const int wave = (int)(blockIdx.x * (blockDim.x >> 5) + (threadIdx.x >> 5));
  const int i0 = wave * 16;
  if (i0 >= nq) return;  // wave-uniform: EXEC stays all-1s for WMMA

  const int l15 = lane & 15;
  const bool hi = lane >= 16;

  // ---- Query tile: load one point per 16-lane half, build A registers ----
  int qrow = i0 + l15;
  if (qrow >= nq) qrow = nq - 1;             // clamp (pad rows masked at end)
  const float qx = Q[qrow * 3 + 0];
  const float qy = Q[qrow * 3 + 1];
  const float qz = Q[qrow * 3 + 2];
  const float qnorm = qx * qx + qy * qy + qz * qz;

  // Pre-scale A by -2 so the WMMA produces the -2*dot cross term directly.
  v2f a;
  a.x = -2.0f * (hi ? qz : qx);
  a.y = hi ? 0.0f : (-2.0f * qy);

  // Row norms for the D layout: VGPR v needs norm of row v (lo) / v+8 (hi).
  float qn[8];
#pragma unroll
  for (int v = 0; v < 8; ++v)
    qn[v] = __shfl(qnorm, v + (hi ? 8 : 0), 32);

  float rmin[8];
  int ridx[8];
#pragma unroll
  for (int v = 0; v < 8; ++v) {
    rmin[v] = 3.0e38f;
    ridx[v] = 0;
  }

  // ---- Scan target points in 16-wide tiles ----
#pragma unroll 2
  for (int j0 = 0; j0 < nt; j0 += 16) {
    int trow = j0 + l15;
    const bool pad = trow >= nt;
    if (pad) trow = nt - 1;
    const float tx = T[trow * 3 + 0];
    const float ty = T[trow * 3 + 1];
    const float tz = T[trow * 3 + 2];
    // Padded columns get a huge norm so they never win the min.
    const float tn = pad ? 3.0e38f : (tx * tx + ty * ty + tz * tz);

    if (j0 + 16 < nt)  // speculative prefetch of next tile
      __builtin_prefetch(&T[(j0 + 16 + l15) * 3], 0, 0);

    v2f b;
    b.x = hi ? tz : tx;
    b.y = hi ? 0.0f : ty;

    // Seed the accumulator with |q|^2 + |t|^2; WMMA adds -2*dot on top.
    v8f c;
#pragma unroll
    for (int v = 0; v < 8; ++v) c[v] = qn[v] + tn;

    c = __builtin_amdgcn_wmma_f32_16x16x4_f32(
        /*neg_a=*/false, a, /*neg_b=*/false, b,
        /*c_mod=*/(short)0, c, /*reuse_a=*/false, /*reuse_b=*/false);

#pragma unroll
    for (int v = 0; v < 8; ++v) {
      if constexpr (WEIGHT_BY_TARGET) {
        if (c[v] < rmin[v]) {
          rmin[v] = c[v];
          ridx[v] = j0 + l15;
        }
      } else {
        rmin[v] = fminf(rmin[v], c[v]);
      }
    }
  }

  // ---- Cross-lane min/argmin within each 16-lane half (columns -> row min) --
#pragma unroll
  for (int m = 1; m <= 8; m <<= 1) {
#pragma unroll
    for (int v = 0; v < 8; ++v) {
      const float od = __shfl_xor(rmin[v], m, 32);
      if constexpr (WEIGHT_BY_TARGET) {
        const int oi = __shfl_xor(ridx[v], m, 32);
        if (od < rmin[v] || (od == rmin[v] && oi < ridx[v])) {
          rmin[v] = od;
          ridx[v] = oi;
        }
      } else {
        rmin[v] = fminf(rmin[v], od);
      }
    }
  }

  // ---- Weighted per-row sum; lo half holds rows 0-7, hi half rows 8-15 ----
  float partial = 0.0f;
#pragma unroll
  for (int v = 0; v < 8; ++v) {
    const int row = i0 + v + (hi ? 8 : 0);
    if (row < nq) {
      const float d = sqrtf(fmaxf(rmin[v], 0.0f));
      const float w = WEIGHT_BY_TARGET ? prob[ridx[v]] : prob[row];
      partial = fmaf(d, w, partial);
    }
  }
  // Combine the two halves (all lanes in a half carry identical values).
  partial += __shfl_xor(partial, 16, 32);
  if (lane == 0) partials[wave] = partial;
}

// Deterministic fixed-order tree reduction of the per-wave partials.
__global__ void __launch_bounds__(256)
chamfer_reduce_kernel(const float* __restrict__ ws, int n,
                      float* __restrict__ out) {
  __shared__ float s[256];
  float acc = 0.0f;
  for (int i = threadIdx.x; i < n; i += 256) acc += ws[i];
  s[threadIdx.x] = acc;
  __syncthreads();
#pragma unroll
  for (int ofs = 128; ofs > 0; ofs >>= 1) {
    if ((int)threadIdx.x < ofs) s[threadIdx.x] += s[threadIdx.x + ofs];
    __syncthreads();
  }
  if (threadIdx.x == 0) out[0] = s[0];
}

extern "C" void kernel_launch(void* const* d_in, const int* in_sizes, int n_in,
                              void* d_out, int out_size, void* d_ws,
                              size_t ws_size, hipStream_t stream) {
  const float* P = (const float*)d_in[0];     // (N,3)
  const float* Ps = (const float*)d_in[1];    // (M,3)
  const float* prob = (const float*)d_in[2];  // (M,)
  float* out = (float*)d_out;
  float* ws = (float*)d_ws;

  const int N = in_sizes[0] / 3;
  const int M = in_sizes[1] / 3;

  const int nw_o = (N + 15) / 16;  // waves for P -> Ps (argmin weighted)
  const int nw_s = (M + 15) / 16;  // waves for Ps -> P (row weighted)

  // 8 waves (256 threads) per block.
  const int blocks_o = (nw_o + 7) / 8;
  const int blocks_s = (nw_s + 7) / 8;

  // P -> Ps: weight by prob[argmin over Ps]
  chamfer_dir_kernel<true><<<blocks_o, 256, 0, stream>>>(P, N, Ps, M, prob, ws);
  // Ps -> P: weight by prob[query row]
  chamfer_dir_kernel<false><<<blocks_s, 256, 0, stream>>>(Ps, M, P, N, prob,
                                                          ws + nw_o);
  chamfer_reduce_kernel<<<1, 256, 0, stream>>>(ws, nw_o + nw_s, out);
}